// Filter_67740224192698
// MI455X (gfx1250) — compile-verified
//
#include <hip/hip_runtime.h>
#include <math.h>

#define NN 10000
#define NE 320000
#define DIM 128
#define NL 3

typedef __attribute__((ext_vector_type(2))) float v2f;
typedef __attribute__((ext_vector_type(8))) float v8f;

// ---------------------------------------------------------------------------
// GEMM: out = X[N,128] @ W[128,128] + b, via V_WMMA_F32_16X16X4_F32.
// grid.x = N/16 (M tile), grid.y = 0..3 selects (Wq,Wk,Wv,Wskip).
// block = 256 threads = 8 waves; wave w owns output columns [16w,16w+16).
// A frag (16x4 f32): lane L, vgpr j -> (row = L%16, k = k0 + 2*(L/16) + j)
// B frag (4x16 f32): lane L, vgpr j -> (k  = k0 + 2*(L/16) + j, col = L%16)
// C/D (16x16 f32):   vgpr r, lane L -> (row = r + 8*(L/16),   col = L%16)
// ---------------------------------------------------------------------------
__global__ __launch_bounds__(256) void tc_gemm4(
    const float* __restrict__ X,
    const float* __restrict__ Wq, const float* __restrict__ bq,
    const float* __restrict__ Wk, const float* __restrict__ bk,
    const float* __restrict__ Wv, const float* __restrict__ bv,
    const float* __restrict__ Ws, const float* __restrict__ bs,
    float* __restrict__ Q, float* __restrict__ Kb, float* __restrict__ Vb,
    float* __restrict__ Stot)
{
    const int wave = threadIdx.x >> 5;
    const int lane = threadIdx.x & 31;
    const int m0 = blockIdx.x * 16;
    const int n0 = wave * 16;
    const int which = blockIdx.y;

    const float* W; const float* bias;
    switch (which) {
        case 0:  W = Wq; bias = bq; break;
        case 1:  W = Wk; bias = bk; break;
        case 2:  W = Wv; bias = bv; break;
        default: W = Ws; bias = bs; break;
    }

    const int lrow = lane & 15;
    const int lhi  = (lane >> 4) << 1;           // 0 or 2
    const float* arow = X + (size_t)(m0 + lrow) * DIM;
    const float* bcol = W + n0 + lrow;

    v8f acc = {};
    #pragma unroll
    for (int k0 = 0; k0 < DIM; k0 += 4) {
        const int ka = k0 + lhi;                  // even -> 8B aligned
        const v2f a = *(const v2f*)(arow + ka);   // global_load_b64
        v2f b;
        b.x = bcol[(size_t)ka * DIM];
        b.y = bcol[(size_t)(ka + 1) * DIM];
        acc = __builtin_amdgcn_wmma_f32_16x16x4_f32(
            false, a, false, b, (short)0, acc, false, false);
    }

    const int col = n0 + lrow;
    const float bb = bias[col];
    const int rbase = m0 + ((lane >> 4) << 3);
    #pragma unroll
    for (int r = 0; r < 8; ++r) {
        const size_t idx = (size_t)(rbase + r) * DIM + col;
        const float val = acc[r] + bb;
        if      (which == 0) Q[idx]  = val;
        else if (which == 1) Kb[idx] = val;
        else if (which == 2) Vb[idx] = val;
        else                 Stot[idx] += val;    // skip projection into s accum
    }
}

// ---------------------------------------------------------------------------
// small helpers
// ---------------------------------------------------------------------------
__global__ void tc_zero(float* p, int n) {
    int i = blockIdx.x * blockDim.x + threadIdx.x;
    if (i < n) p[i] = 0.0f;
}
__global__ void tc_copy(float* __restrict__ d, const float* __restrict__ s, int n) {
    int i = blockIdx.x * blockDim.x + threadIdx.x;
    if (i < n) d[i] = s[i];
}
__global__ void tc_softmax_init(unsigned* __restrict__ amaxk, float* __restrict__ den, int n) {
    int i = blockIdx.x * blockDim.x + threadIdx.x;
    if (i < n) { amaxk[i] = 0u; den[i] = 0.0f; }   // key 0 == -inf under mapping
}

__device__ __forceinline__ unsigned f2key(float f) {
    unsigned u = __float_as_uint(f);
    return (u & 0x80000000u) ? ~u : (u | 0x80000000u);
}
__device__ __forceinline__ float key2f(unsigned k) {
    return __uint_as_float((k & 0x80000000u) ? (k & 0x7FFFFFFFu) : ~k);
}

// ---------------------------------------------------------------------------
// alpha[e] = dot(q[dst], k[src] + edge_emb) / sqrt(D); segment-max via
// order-preserving uint atomicMax. One wave32 per edge, float4 per lane.
// ---------------------------------------------------------------------------
__global__ __launch_bounds__(256) void tc_alpha(
    const int* __restrict__ src, const int* __restrict__ dst,
    const float* __restrict__ ea, const float* __restrict__ We,
    const float* __restrict__ Q, const float* __restrict__ Kb,
    float* __restrict__ alpha, unsigned* __restrict__ amaxk)
{
    const int lane = threadIdx.x & 31;
    const int e = (int)((blockIdx.x * blockDim.x + threadIdx.x) >> 5);
    const int s = src[e], d = dst[e];
    const float ea0 = ea[2 * e], ea1 = ea[2 * e + 1];
    const int c0 = lane << 2;

    const float4 qv = *(const float4*)(Q  + (size_t)d * DIM + c0);
    const float4 kv = *(const float4*)(Kb + (size_t)s * DIM + c0);
    const float4 w0 = *(const float4*)(We + c0);
    const float4 w1 = *(const float4*)(We + DIM + c0);

    float dot = qv.x * (kv.x + ea0 * w0.x + ea1 * w1.x)
              + qv.y * (kv.y + ea0 * w0.y + ea1 * w1.y)
              + qv.z * (kv.z + ea0 * w0.z + ea1 * w1.z)
              + qv.w * (kv.w + ea0 * w0.w + ea1 * w1.w);
    #pragma unroll
    for (int off = 16; off > 0; off >>= 1) dot += __shfl_xor(dot, off, 32);
    dot *= 0.08838834764831845f;                   // 1/sqrt(128)

    if (lane == 0) {
        alpha[e] = dot;
        atomicMax(amaxk + d, f2key(dot));
    }
}

// ex[e] = exp(alpha - amax[dst]); denom[dst] += ex  (thread per edge)
__global__ void tc_exden(
    const int* __restrict__ dst, const float* __restrict__ alpha,
    const unsigned* __restrict__ amaxk,
    float* __restrict__ ex, float* __restrict__ den)
{
    int e = blockIdx.x * blockDim.x + threadIdx.x;
    if (e >= NE) return;
    const int d = dst[e];
    const float v = expf(alpha[e] - key2f(amaxk[d]));
    ex[e] = v;
    atomicAdd(den + d, v);
}

// Stot[dst] += (v[src] + edge_emb) * (ex/denom[dst]) — wave per edge, f32 atomics
__global__ __launch_bounds__(256) void tc_agg(
    const int* __restrict__ src, const int* __restrict__ dst,
    const float* __restrict__ ea, const float* __restrict__ We,
    const float* __restrict__ Vb, const float* __restrict__ ex,
    const float* __restrict__ den, float* __restrict__ Stot)
{
    const int lane = threadIdx.x & 31;
    const int e = (int)((blockIdx.x * blockDim.x + threadIdx.x) >> 5);
    const int s = src[e], d = dst[e];
    const float ea0 = ea[2 * e], ea1 = ea[2 * e + 1];
    const float w = ex[e] / den[d];
    const int c0 = lane << 2;

    const float4 vv = *(const float4*)(Vb + (size_t)s * DIM + c0);
    const float4 w0 = *(const float4*)(We + c0);
    const float4 w1 = *(const float4*)(We + DIM + c0);
    float* o = Stot + (size_t)d * DIM + c0;
    atomicAdd(o + 0, w * (vv.x + ea0 * w0.x + ea1 * w1.x));
    atomicAdd(o + 1, w * (vv.y + ea0 * w0.y + ea1 * w1.y));
    atomicAdd(o + 2, w * (vv.z + ea0 * w0.z + ea1 * w1.z));
    atomicAdd(o + 3, w * (vv.w + ea0 * w0.w + ea1 * w1.w));
}

// sparse A@x : xn[src[e]] += x[dst[e]]  (wave per edge)
__global__ __launch_bounds__(256) void tc_spmm(
    const int* __restrict__ src, const int* __restrict__ dst,
    const float* __restrict__ x, float* __restrict__ xn)
{
    const int lane = threadIdx.x & 31;
    const int e = (int)((blockIdx.x * blockDim.x + threadIdx.x) >> 5);
    const int s = src[e], d = dst[e];
    const int c0 = lane << 2;
    const float4 xv = *(const float4*)(x + (size_t)d * DIM + c0);
    float* o = xn + (size_t)s * DIM + c0;
    atomicAdd(o + 0, xv.x);
    atomicAdd(o + 1, xv.y);
    atomicAdd(o + 2, xv.z);
    atomicAdd(o + 3, xv.w);
}

__global__ void tc_sigmoid(const float* __restrict__ s, float* __restrict__ out, int n) {
    int i = blockIdx.x * blockDim.x + threadIdx.x;
    if (i < n) out[i] = 1.0f / (1.0f + expf(-s[i]));
}

// ---------------------------------------------------------------------------
extern "C" void kernel_launch(void* const* d_in, const int* in_sizes, int n_in,
                              void* d_out, int out_size, void* d_ws, size_t ws_size,
                              hipStream_t stream) {
    (void)in_sizes; (void)n_in; (void)out_size; (void)ws_size;
    const float* X    = (const float*)d_in[0];
    const int*   ei   = (const int*)d_in[1];
    const float* ea   = (const float*)d_in[2];
    const float* Wq   = (const float*)d_in[3];
    const float* bq   = (const float*)d_in[4];
    const float* Wk   = (const float*)d_in[5];
    const float* bk   = (const float*)d_in[6];
    const float* Wv   = (const float*)d_in[7];
    const float* bv   = (const float*)d_in[8];
    const float* We   = (const float*)d_in[9];
    const float* Wsk  = (const float*)d_in[10];
    const float* bsk  = (const float*)d_in[11];
    const int* srcp = ei;
    const int* dstp = ei + NE;
    float* out = (float*)d_out;

    // workspace carve-up (all 256B aligned): ~34 MB total
    const size_t ND = (size_t)NN * DIM;
    char* ws = (char*)d_ws;
    size_t off = 0;
    auto alloc = [&](size_t bytes) { void* p = ws + off; off = (off + bytes + 255) & ~(size_t)255; return p; };
    float*    Q     = (float*)alloc(ND * 4);
    float*    Kb    = (float*)alloc(ND * 4);
    float*    Vb    = (float*)alloc(ND * 4);
    float*    Stot  = (float*)alloc(ND * 4);
    float*    xA    = (float*)alloc(ND * 4);
    float*    xB    = (float*)alloc(ND * 4);
    float*    alpha = (float*)alloc((size_t)NE * 4);
    float*    exv   = (float*)alloc((size_t)NE * 4);
    float*    den   = (float*)alloc((size_t)NN * 4);
    unsigned* amaxk = (unsigned*)alloc((size_t)NN * 4);

    const int ndi = (int)ND;
    const int gND = (ndi + 255) / 256;
    const int gN  = (NN + 255) / 256;
    const int gEw = NE / 8;            // wave-per-edge: 8 waves per 256-thr block
    const int gEt = (NE + 255) / 256;  // thread-per-edge

    tc_copy<<<gND, 256, 0, stream>>>(xA, X, ndi);
    tc_zero<<<gND, 256, 0, stream>>>(Stot, ndi);

    float* xcur = xA;
    float* xnext = xB;
    for (int i = 0; i < NL; ++i) {
        const float* We_l = We + (size_t)i * 2 * DIM;
        tc_gemm4<<<dim3(NN / 16, 4), 256, 0, stream>>>(
            xcur,
            Wq  + (size_t)i * DIM * DIM, bq  + (size_t)i * DIM,
            Wk  + (size_t)i * DIM * DIM, bk  + (size_t)i * DIM,
            Wv  + (size_t)i * DIM * DIM, bv  + (size_t)i * DIM,
            Wsk + (size_t)i * DIM * DIM, bsk + (size_t)i * DIM,
            Q, Kb, Vb, Stot);
        tc_softmax_init<<<gN, 256, 0, stream>>>(amaxk, den, NN);
        tc_alpha<<<gEw, 256, 0, stream>>>(srcp, dstp, ea, We_l, Q, Kb, alpha, amaxk);
        tc_exden<<<gEt, 256, 0, stream>>>(dstp, alpha, amaxk, exv, den);
        tc_agg<<<gEw, 256, 0, stream>>>(srcp, dstp, ea, We_l, Vb, exv, den, Stot);
        if (i + 1 < NL) {
            tc_zero<<<gND, 256, 0, stream>>>(xnext, ndi);
            tc_spmm<<<gEw, 256, 0, stream>>>(srcp, dstp, xcur, xnext);
            float* t = xcur; xcur = xnext; xnext = t;
        }
    }
    tc_sigmoid<<<gND, 256, 0, stream>>>(Stot, out, ndi);
}